// Blurring_25125558681617
// MI455X (gfx1250) — compile-verified
//
#include <hip/hip_runtime.h>
#include <stdint.h>

// Median 3x3, REFLECT padding, NHWC f32 (32,224,224,3).
// HBM-bound: stage rows once via CDNA5 async global->LDS DMA (ASYNCcnt),
// rolling 3-row register window + 19-op median network, NT streaming stores.

#define BATCH    32
#define HH       224
#define WW       224
#define CC       3
#define ROWF     (WW * CC)        // 672 floats per image row
#define TILE_H   8
#define LROWS    (TILE_H + 2)     // 10 staged rows (with halo)
#define NSEG     (ROWF / 4)       // 168 float4 segments per row
#define NTHREADS 224              // 7 wave32; 672 = 3 * 224

__device__ __forceinline__ int reflect_idx(int i, int n) {
    // jnp.pad mode="reflect": -1 -> 1, n -> n-2
    if (i < 0)  return -i;
    if (i >= n) return 2 * n - 2 - i;
    return i;
}

__device__ __forceinline__ void cswap(float& a, float& b) {
    float lo = fminf(a, b);
    float hi = fmaxf(a, b);
    a = lo;
    b = hi;
}

__global__ __launch_bounds__(NTHREADS)
void median3x3_kernel(const float* __restrict__ in, float* __restrict__ out) {
    __shared__ float smem[LROWS * ROWF];   // 26880 bytes

    const int tid = threadIdx.x;
    const int b   = blockIdx.x / (HH / TILE_H);
    const int h0  = (blockIdx.x % (HH / TILE_H)) * TILE_H;

    const float* img = in + (size_t)b * HH * ROWF;

    // ---- Stage LROWS rows into LDS with async global->LDS B128 DMA ----
    const uint32_t lds_base =
        (uint32_t)(uintptr_t)(&smem[0]);   // low 32 bits of flat shared ptr = LDS offset
    for (int t = tid; t < LROWS * NSEG; t += NTHREADS) {
        const int lr  = t / NSEG;
        const int seg = t - lr * NSEG;
        const int gr  = reflect_idx(h0 - 1 + lr, HH);
        const uint64_t gaddr =
            (uint64_t)(uintptr_t)(img + (size_t)gr * ROWF + (size_t)seg * 4);
        const uint32_t laddr =
            lds_base + (uint32_t)((lr * ROWF + seg * 4) * sizeof(float));
        asm volatile("global_load_async_to_lds_b128 %0, %1, off"
                     :: "v"(laddr), "v"(gaddr)
                     : "memory");
    }
    asm volatile("s_wait_asynccnt 0" ::: "memory");
    __syncthreads();

    // ---- Median-of-9: rolling 3-row register window, no div/mod ----
    float* o = out + ((size_t)b * HH + h0) * ROWF;

    #pragma unroll
    for (int j = 0; j < 3; ++j) {
        const int x  = tid + j * NTHREADS;                 // 0..671, exact
        // column REFLECT in flat-row terms: w=0 -> w=1 is +3; w=223 -> w=222 is -3
        const int xm = (x < CC)        ? x + CC : x - CC;
        const int xp = (x >= ROWF - CC) ? x - CC : x + CC;

        // preload rows 0 and 1 of the window
        float a0 = smem[xm],        a1 = smem[x],        a2 = smem[xp];
        float b0 = smem[ROWF + xm], b1 = smem[ROWF + x], b2 = smem[ROWF + xp];

        #pragma unroll
        for (int r = 0; r < TILE_H; ++r) {
            const float* rw = &smem[(r + 2) * ROWF];
            const float c0 = rw[xm], c1 = rw[x], c2 = rw[xp];

            float p0 = a0, p1 = a1, p2 = a2;
            float p3 = b0, p4 = b1, p5 = b2;
            float p6 = c0, p7 = c1, p8 = c2;

            cswap(p1, p2); cswap(p4, p5); cswap(p7, p8);
            cswap(p0, p1); cswap(p3, p4); cswap(p6, p7);
            cswap(p1, p2); cswap(p4, p5); cswap(p7, p8);
            cswap(p0, p3); cswap(p5, p8); cswap(p4, p7);
            cswap(p3, p6); cswap(p1, p4); cswap(p2, p5);
            cswap(p4, p7); cswap(p4, p2); cswap(p6, p4);
            cswap(p4, p2);

            __builtin_nontemporal_store(p4, &o[(size_t)r * ROWF + x]);

            a0 = b0; a1 = b1; a2 = b2;   // roll the window
            b0 = c0; b1 = c1; b2 = c2;
        }
    }
}

extern "C" void kernel_launch(void* const* d_in, const int* in_sizes, int n_in,
                              void* d_out, int out_size, void* d_ws, size_t ws_size,
                              hipStream_t stream) {
    const float* in  = (const float*)d_in[0];
    float*       out = (float*)d_out;
    (void)in_sizes; (void)n_in; (void)out_size; (void)d_ws; (void)ws_size;

    dim3 grid(BATCH * (HH / TILE_H));   // 896 blocks
    dim3 block(NTHREADS);
    median3x3_kernel<<<grid, block, 0, stream>>>(in, out);
}